// PhraseAnalyzer_45603962749789
// MI455X (gfx1250) — compile-verified
//
#include <hip/hip_runtime.h>
#include <hip/hip_bf16.h>

typedef __attribute__((ext_vector_type(2))) float v2f;
typedef __attribute__((ext_vector_type(8))) float v8f;

#define B_ 512
#define S_ 2048
#define T_ 31
#define TP 32   // padded tag dimension for WMMA K/N
#define LN2 0.69314718056f

__device__ __forceinline__ float lane_max16(float v) {
#pragma unroll
  for (int off = 1; off < 16; off <<= 1)
    v = fmaxf(v, __shfl_xor(v, off, 32));
  return v;
}
__device__ __forceinline__ float lane_sum16(float v) {
#pragma unroll
  for (int off = 1; off < 16; off <<= 1)
    v += __shfl_xor(v, off, 32);
  return v;
}

// One async Global->LDS B32 copy (per lane). ASYNCcnt-tracked.
__device__ __forceinline__ void async_copy_b32(const float* gp, float* lp) {
#if __has_builtin(__builtin_amdgcn_global_load_async_to_lds_b32)
  __builtin_amdgcn_global_load_async_to_lds_b32(
      (__attribute__((address_space(1))) int*)const_cast<float*>(gp),
      (__attribute__((address_space(3))) int*)lp, 0, 0);
#else
  unsigned lo = (unsigned)(unsigned long)(__attribute__((address_space(3))) void*)lp;
  asm volatile("global_load_async_to_lds_b32 %0, %1, off"
               :: "v"(lo), "v"(gp) : "memory");
#endif
}

// ---------------------------------------------------------------------------
// Forward algorithm (denominator) via WMMA f32 16x16x4.
// One wave handles 16 batches; alpha kept in WMMA C/D register layout
// (M = batch, N = tag). Contraction: exp(alpha-rowmax) x exp(trans) on the
// matrix pipe; emissions double-buffered through LDS with async copies.
// ---------------------------------------------------------------------------
__global__ __launch_bounds__(128)
void crf_forward_denominator(const float* __restrict__ em,
                             const unsigned char* __restrict__ mask,
                             const float* __restrict__ startt,
                             const float* __restrict__ endt,
                             const float* __restrict__ trans,
                             float* __restrict__ den) {
  __shared__ float et_cm[TP * TP];            // exp(trans), column-major [n][k]
  __shared__ float ascr_all[4][16 * TP];      // per-wave transpose scratch
  __shared__ float ebuf_all[4][2][16 * TP];   // per-wave emission double buffer

  const int tid = threadIdx.x;
  // exp(transitions), zero-padded; column-major so B-operand pairs are b64 loads
  for (int idx = tid; idx < TP * TP; idx += 128) {
    int n = idx >> 5, k = idx & 31;
    float v = 0.0f;
    if (n < T_ && k < T_) v = __expf(trans[k * T_ + n]);
    et_cm[idx] = v;
  }
  __syncthreads();

  const int wave = tid >> 5;
  const int lane = tid & 31;
  const int half = lane >> 4;
  const int l16  = lane & 15;
  float* ascr = ascr_all[wave];
  const int wbase = (blockIdx.x * 4 + wave) * 16;  // first batch of this wave

  // Preload B operand (exp_trans) tiles: 2 N-tiles x 8 K-blocks, 2 VGPRs each.
  // B layout for 16x16x4: VGPR0 = rows K,K+2 ; VGPR1 = rows K+1,K+3 ; N across lanes.
  v2f btile0[8], btile1[8];
#pragma unroll
  for (int kb = 0; kb < 8; ++kb) {
    int k = 4 * kb + 2 * half;
    btile0[kb] = *(const v2f*)&et_cm[(l16) * TP + k];
    btile1[kb] = *(const v2f*)&et_cm[(16 + l16) * TP + k];
  }

  // alpha_0 = start + emissions[:,0,:], held in D layout:
  // c{0,1}[r] -> batch m = r + 8*half, tag col = {l16, 16+l16}
  v8f c0, c1;
#pragma unroll
  for (int r = 0; r < 8; ++r) {
    long bg = wbase + r + 8 * half;
    int col1 = 16 + l16;
    int c1c = col1 > (T_ - 1) ? (T_ - 1) : col1;   // col 31 is dead (masked below)
    c0[r] = startt[l16] + em[(bg * (long)S_ + 0) * T_ + l16];
    c1[r] = startt[c1c] + em[(bg * (long)S_ + 0) * T_ + c1c];
  }

  // Prime the pipeline: async-stage emissions for step 1 into buffer 1.
  {
    float* eb = ebuf_all[wave][1];
#pragma unroll
    for (int r = 0; r < 8; ++r) {
      long bg = wbase + r + 8 * half;
      int m = r + 8 * half;
      int c1c = (16 + l16) > (T_ - 1) ? (T_ - 1) : (16 + l16);
      async_copy_b32(&em[(bg * (long)S_ + 1) * T_ + l16], &eb[m * TP + l16]);
      async_copy_b32(&em[(bg * (long)S_ + 1) * T_ + c1c], &eb[m * TP + 16 + l16]);
    }
  }

  for (int s = 1; s < S_; ++s) {
    // Issue async copies for step s+1 while we compute step s, then wait for
    // the current buffer only (ASYNCcnt decrements in order -> <=16 means the
    // 16 loads issued last iteration are complete, next 16 stay in flight).
    if (s + 1 < S_) {
      float* eb = ebuf_all[wave][(s + 1) & 1];
#pragma unroll
      for (int r = 0; r < 8; ++r) {
        long bg = wbase + r + 8 * half;
        int m = r + 8 * half;
        int c1c = (16 + l16) > (T_ - 1) ? (T_ - 1) : (16 + l16);
        async_copy_b32(&em[(bg * (long)S_ + (s + 1)) * T_ + l16], &eb[m * TP + l16]);
        async_copy_b32(&em[(bg * (long)S_ + (s + 1)) * T_ + c1c], &eb[m * TP + 16 + l16]);
      }
      asm volatile("s_wait_asynccnt 0x10" ::: "memory");
    } else {
      asm volatile("s_wait_asynccnt 0x0" ::: "memory");
    }

    // per-batch row max (mask the padded column 31)
    float rmax[8];
#pragma unroll
    for (int r = 0; r < 8; ++r) {
      float v1 = (l16 == 15) ? -3.0e38f : c1[r];
      rmax[r] = lane_max16(fmaxf(c0[r], v1));
    }

    // exp(alpha - rowmax) -> LDS in A (batch-major) layout for the transpose
#pragma unroll
    for (int r = 0; r < 8; ++r) {
      int m = r + 8 * half;
      float e0 = __expf(c0[r] - rmax[r]);
      float e1 = (l16 == 15) ? 0.0f : __expf(c1[r] - rmax[r]);
      ascr[m * TP + l16]      = e0;
      ascr[m * TP + 16 + l16] = e1;
    }
    asm volatile("s_wait_dscnt 0x0" ::: "memory");

    // p' = exp_alpha (16x32) x exp_trans (32x32) using 16 accumulating WMMAs.
    // A layout for 16x16x4: M across lanes, VGPR0 = K,K+2 ; VGPR1 = K+1,K+3.
    v8f d0 = {0.f, 0.f, 0.f, 0.f, 0.f, 0.f, 0.f, 0.f};
    v8f d1 = {0.f, 0.f, 0.f, 0.f, 0.f, 0.f, 0.f, 0.f};
#pragma unroll
    for (int kb = 0; kb < 8; ++kb) {
      v2f a = *(const v2f*)&ascr[l16 * TP + 4 * kb + 2 * half];
      d0 = __builtin_amdgcn_wmma_f32_16x16x4_f32(false, a, false, btile0[kb],
                                                 (short)0, d0, false, false);
      d1 = __builtin_amdgcn_wmma_f32_16x16x4_f32(false, a, false, btile1[kb],
                                                 (short)0, d1, false, false);
    }

    // alpha' = rowmax + log(p') + emission ; keep old alpha where mask==0.
    // Sums are clamped to a *normal* float, so raw v_log_f32 (base-2) is safe.
    const float* eb = ebuf_all[wave][s & 1];
#pragma unroll
    for (int r = 0; r < 8; ++r) {
      long bg = wbase + r + 8 * half;
      int m = r + 8 * half;
      bool mk = mask[bg * (long)S_ + s] != 0;
      float e0v = eb[m * TP + l16];
      float e1v = eb[m * TP + 16 + l16];
      float n0 = rmax[r] + LN2 * __builtin_amdgcn_logf(fmaxf(d0[r], 1e-37f)) + e0v;
      float n1 = rmax[r] + LN2 * __builtin_amdgcn_logf(fmaxf(d1[r], 1e-37f)) + e1v;
      c0[r] = mk ? n0 : c0[r];
      c1[r] = mk ? n1 : c1[r];
    }
  }

  // denominator[b] = logsumexp_j(alpha[b,j] + end[j])
#pragma unroll
  for (int r = 0; r < 8; ++r) {
    int col1 = 16 + l16;
    int c1c = col1 > (T_ - 1) ? (T_ - 1) : col1;
    float v0 = c0[r] + endt[l16];
    float v1 = (l16 == 15) ? -3.0e38f : c1[r] + endt[c1c];
    float mx = lane_max16(fmaxf(v0, v1));
    float sm = __expf(v0 - mx) + ((l16 == 15) ? 0.0f : __expf(v1 - mx));
    sm = lane_sum16(sm);
    float dn = mx + LN2 * __builtin_amdgcn_logf(fmaxf(sm, 1e-37f));
    if (l16 == 0) den[wbase + r + 8 * half] = dn;
  }
}

// ---------------------------------------------------------------------------
// Numerator: path score per batch + mask count + last-tag end transition.
// ---------------------------------------------------------------------------
__global__ __launch_bounds__(256)
void crf_numerator(const float* __restrict__ em,
                   const int* __restrict__ tags,
                   const unsigned char* __restrict__ mask,
                   const float* __restrict__ startt,
                   const float* __restrict__ endt,
                   const float* __restrict__ trans,
                   float* __restrict__ num, float* __restrict__ msum) {
  __shared__ float ssum[256];
  __shared__ float scnt[256];
  __shared__ int   smax[256];
  const int b = blockIdx.x;
  const int tid = threadIdx.x;
  float part = 0.f, cnt = 0.f;
  int send = 0;
  for (int s = tid; s < S_; s += 256) {
    int tg = tags[b * S_ + s];
    unsigned char mk = mask[b * S_ + s];
    float e = em[((long)b * S_ + s) * T_ + tg];
    if (s == 0) {
      part += startt[tg] + e;
    } else if (mk) {
      int tp = tags[b * S_ + s - 1];
      part += trans[tp * T_ + tg] + e;
    }
    if (mk) { cnt += 1.f; send = max(send, s); }
  }
  ssum[tid] = part; scnt[tid] = cnt; smax[tid] = send;
  __syncthreads();
  for (int off = 128; off > 0; off >>= 1) {
    if (tid < off) {
      ssum[tid] += ssum[tid + off];
      scnt[tid] += scnt[tid + off];
      smax[tid] = max(smax[tid], smax[tid + off]);
    }
    __syncthreads();
  }
  if (tid == 0) {
    int lt = tags[b * S_ + smax[0]];
    num[b]  = ssum[0] + endt[lt];
    msum[b] = scnt[0];
  }
}

// ---------------------------------------------------------------------------
// Final scalar: sum(num - den) / sum(mask)
// ---------------------------------------------------------------------------
__global__ __launch_bounds__(256)
void crf_finalize(const float* __restrict__ num, const float* __restrict__ den,
                  const float* __restrict__ msum, float* __restrict__ out) {
  __shared__ float sl[256], sm[256];
  int tid = threadIdx.x;
  float l = 0.f, m = 0.f;
  for (int i = tid; i < B_; i += 256) { l += num[i] - den[i]; m += msum[i]; }
  sl[tid] = l; sm[tid] = m;
  __syncthreads();
  for (int off = 128; off > 0; off >>= 1) {
    if (tid < off) { sl[tid] += sl[tid + off]; sm[tid] += sm[tid + off]; }
    __syncthreads();
  }
  if (tid == 0) out[0] = sl[0] / sm[0];
}

extern "C" void kernel_launch(void* const* d_in, const int* in_sizes, int n_in,
                              void* d_out, int out_size, void* d_ws, size_t ws_size,
                              hipStream_t stream) {
  const float*         em     = (const float*)d_in[0];
  const int*           tags   = (const int*)d_in[1];
  const unsigned char* mask   = (const unsigned char*)d_in[2];
  const float*         startt = (const float*)d_in[3];
  const float*         endt   = (const float*)d_in[4];
  const float*         trans  = (const float*)d_in[5];

  float* num  = (float*)d_ws;
  float* den  = num + B_;
  float* msum = den + B_;

  crf_numerator<<<B_, 256, 0, stream>>>(em, tags, mask, startt, endt, trans, num, msum);
  crf_forward_denominator<<<B_ / 64, 128, 0, stream>>>(em, mask, startt, endt, trans, den);
  crf_finalize<<<1, 256, 0, stream>>>(num, den, msum, (float*)d_out);
}